// Qwen3OmniMoeExperts_26663156973647
// MI455X (gfx1250) — compile-verified
//
#include <hip/hip_runtime.h>
#include <math.h>

// Problem constants (match reference)
#define T_TOK  1024
#define K_TOP  2
#define E_EXP  8
#define H_DIM  2048
#define I_DIM  1024
#define P_PAIR (T_TOK * K_TOP)   // 2048 (token, slot) pairs

typedef float v2f __attribute__((ext_vector_type(2)));
typedef float v8f __attribute__((ext_vector_type(8)));

// K-panel of 32 staged in LDS; stride padded 32->36 floats (144B):
//  - 16B alignment for b128 staging stores, 8B for b64 frag reads
//  - 36*i mod 64 = 4*(9*i mod 16): 16 distinct banks per lane half; the
//    upper-K half (+2 floats) lands on disjoint bank pairs -> conflict-free.
#define KP         32
#define LDS_STRIDE 36

// ---------------- routing ----------------

__global__ void moe_route_init(int* counts, int* fill) {
    int i = threadIdx.x;
    if (i < E_EXP) { counts[i] = 0; fill[i] = 0; }
}

__global__ void moe_route_count(const int* __restrict__ sel, int* counts) {
    int p = blockIdx.x * blockDim.x + threadIdx.x;
    if (p < P_PAIR) atomicAdd(&counts[sel[p]], 1);
}

__global__ void moe_route_scan(const int* __restrict__ counts, int* offsets) {
    if (threadIdx.x == 0) {
        int acc = 0;
        for (int e = 0; e < E_EXP; ++e) { offsets[e] = acc; acc += counts[e]; }
    }
}

__global__ void moe_route_fill(const int* __restrict__ sel,
                               const int* __restrict__ offsets,
                               int* fill, int* rows) {
    int p = blockIdx.x * blockDim.x + threadIdx.x;
    if (p < P_PAIR) {
        int e = sel[p];
        int pos = offsets[e] + atomicAdd(&fill[e], 1);
        rows[pos] = p;                 // pair id = t*K + k
    }
}

#define WMMA_F32(A, B, C) \
    __builtin_amdgcn_wmma_f32_16x16x4_f32(false, (A), false, (B), (short)0, (C), false, false)

// ---------------- GEMM1 + SiLU*up (fused gate/up) ----------------
// grid: (m_tiles=64 of 32 rows, I/128=8, E).  block: 256 = 8 waves.
// Each wave: 32x16 gate tile + 32x16 up tile (4 accumulators), K = H = 2048.

__global__ __launch_bounds__(256)
void moe_gemm1_silu(const float* __restrict__ hidden,
                    const float* __restrict__ gup,     // (E, 2I, H)
                    const int*   __restrict__ counts,
                    const int*   __restrict__ offsets,
                    const int*   __restrict__ rows,
                    float*       __restrict__ inter) { // (P, I) compacted
    const int e     = blockIdx.z;
    const int mtile = blockIdx.x;
    const int cnt   = counts[e];
    const int valid = cnt - mtile * 32;
    if (valid <= 0) return;                        // block-uniform exit
    const int off   = offsets[e];
    const int nBase = blockIdx.y * 128;

    __shared__ float sA [32  * LDS_STRIDE];
    __shared__ float sBg[128 * LDS_STRIDE];
    __shared__ float sBu[128 * LDS_STRIDE];

    const int tid  = threadIdx.x;
    const int lane = tid & 31;
    const int wv   = tid >> 5;       // 8 waves -> 8 n-subtiles of 16
    const int half = lane >> 4;      // K-half select (lanes 16-31 = upper K)
    const int lm   = lane & 15;      // M (for A) / N (for B) index

    const size_t gupBase = (size_t)e * (2 * I_DIM) * H_DIM;
    const float* wg = gup + gupBase + (size_t)nBase * H_DIM;           // gate rows
    const float* wu = gup + gupBase + (size_t)(I_DIM + nBase) * H_DIM; // up rows

    // ---- fixed per-thread staging addresses (one b128 per matrix-chunk) ----
    const int rB = tid >> 3;         // 0..31
    const int c4 = tid & 7;          // 0..7 (16B column)
    // A: rows clamped into range -- garbage rows only feed masked C rows.
    int rr = mtile * 32 + rB; if (rr >= cnt) rr = cnt - 1;
    const float* aSrc = hidden + (size_t)(rows[off + rr] / K_TOP) * H_DIM + c4 * 4;
    float* aDst = &sA[rB * LDS_STRIDE + c4 * 4];
    const float* gSrc = wg + (size_t)rB * H_DIM + c4 * 4;  // +j*32 rows by const offset
    const float* uSrc = wu + (size_t)rB * H_DIM + c4 * 4;
    float* gDst = &sBg[rB * LDS_STRIDE + c4 * 4];
    float* uDst = &sBu[rB * LDS_STRIDE + c4 * 4];

    // ---- fixed fragment read addresses ----
    const float* a0B = &sA [ lm            * LDS_STRIDE + half * 2];
    const float* a1B = a0B + 16 * LDS_STRIDE;
    const float* bgB = &sBg[(wv * 16 + lm) * LDS_STRIDE + half * 2];
    const float* buB = &sBu[(wv * 16 + lm) * LDS_STRIDE + half * 2];

    v8f cg0 = {}, cg1 = {}, cu0 = {}, cu1 = {};

    for (int k0 = 0; k0 < H_DIM; k0 += KP) {
        __syncthreads();
        *(float4*)aDst = *(const float4*)aSrc;
        #pragma unroll
        for (int j = 0; j < 4; ++j) {   // 4 x 32 = 128 rows each
            *(float4*)(gDst + j * 32 * LDS_STRIDE) =
                *(const float4*)(gSrc + (size_t)j * 32 * H_DIM);
            *(float4*)(uDst + j * 32 * LDS_STRIDE) =
                *(const float4*)(uSrc + (size_t)j * 32 * H_DIM);
        }
        aSrc += KP; gSrc += KP; uSrc += KP;
        __syncthreads();

        // prefetch next K-panel's B rows (one 128B line per row) during compute
        if (k0 + KP < H_DIM) {
            const float* pb = (tid < 128) ? wg : wu;
            __builtin_prefetch(pb + (size_t)(tid & 127) * H_DIM + k0 + KP, 0, 3);
        }

        #pragma unroll
        for (int kk = 0; kk < KP; kk += 4) {
            v2f a0 = *(const v2f*)(a0B + kk);
            v2f a1 = *(const v2f*)(a1B + kk);
            v2f bg = *(const v2f*)(bgB + kk);
            v2f bu = *(const v2f*)(buB + kk);
            cg0 = WMMA_F32(a0, bg, cg0);
            cu0 = WMMA_F32(a0, bu, cu0);
            cg1 = WMMA_F32(a1, bg, cg1);
            cu1 = WMMA_F32(a1, bu, cu1);
        }
    }

    // epilogue: inter = silu(gate) * up   (C layout: VGPR r -> M = r + 8*half)
    const int    nCol    = nBase + wv * 16 + lm;
    const size_t rowBase = (size_t)(off + mtile * 32);
    #pragma unroll
    for (int r = 0; r < 8; ++r) {
        int m0 = r + 8 * half;
        if (m0 < valid) {
            float g = cg0[r], u = cu0[r];
            inter[(rowBase + m0) * I_DIM + nCol] = (g / (1.0f + __expf(-g))) * u;
        }
        int m1 = m0 + 16;
        if (m1 < valid) {
            float g = cg1[r], u = cu1[r];
            inter[(rowBase + m1) * I_DIM + nCol] = (g / (1.0f + __expf(-g))) * u;
        }
    }
}

// ---------------- GEMM2 + scatter to output ----------------
// grid: (m_tiles=64 of 32 rows, H/256=8, E).  Each wave: 32x32 tile, K = I.

__global__ __launch_bounds__(256)
void moe_gemm2_scatter(const float* __restrict__ inter,   // (P, I) compacted
                       const float* __restrict__ dproj,   // (E, H, I)
                       const int*   __restrict__ counts,
                       const int*   __restrict__ offsets,
                       const int*   __restrict__ rows,
                       float*       __restrict__ out) {   // (T, K, H)
    const int e     = blockIdx.z;
    const int mtile = blockIdx.x;
    const int cnt   = counts[e];
    const int valid = cnt - mtile * 32;
    if (valid <= 0) return;
    const int off   = offsets[e];
    const int nBase = blockIdx.y * 256;

    __shared__ int   sPair[32];
    __shared__ float sA[32  * LDS_STRIDE];
    __shared__ float sB[256 * LDS_STRIDE];

    const int tid  = threadIdx.x;
    if (tid < 32) {
        int r = mtile * 32 + tid; if (r >= cnt) r = cnt - 1;   // clamped; masked at store
        sPair[tid] = rows[off + r];
    }

    const int lane = tid & 31;
    const int wv   = tid >> 5;
    const int half = lane >> 4;
    const int lm   = lane & 15;

    const float* w = dproj + (size_t)e * H_DIM * I_DIM + (size_t)nBase * I_DIM;

    const int rB = tid >> 3;
    const int c4 = tid & 7;
    int rr = mtile * 32 + rB; if (rr >= cnt) rr = cnt - 1;
    const float* aSrc = inter + (size_t)(off + rr) * I_DIM + c4 * 4;
    float* aDst = &sA[rB * LDS_STRIDE + c4 * 4];
    const float* bSrc = w + (size_t)rB * I_DIM + c4 * 4;     // +j*32 rows by const offset
    float* bDst = &sB[rB * LDS_STRIDE + c4 * 4];

    const float* a0B = &sA[ lm            * LDS_STRIDE + half * 2];
    const float* a1B = a0B + 16 * LDS_STRIDE;
    const float* b0B = &sB[(wv * 32 + lm) * LDS_STRIDE + half * 2];
    const float* b1B = b0B + 16 * LDS_STRIDE;

    v8f c00 = {}, c01 = {}, c10 = {}, c11 = {};

    for (int k0 = 0; k0 < I_DIM; k0 += KP) {
        __syncthreads();
        *(float4*)aDst = *(const float4*)aSrc;
        #pragma unroll
        for (int j = 0; j < 8; ++j) {   // 8 x 32 = 256 weight rows
            *(float4*)(bDst + j * 32 * LDS_STRIDE) =
                *(const float4*)(bSrc + (size_t)j * 32 * I_DIM);
        }
        aSrc += KP; bSrc += KP;
        __syncthreads();

        if (k0 + KP < I_DIM)
            __builtin_prefetch(w + (size_t)tid * I_DIM + k0 + KP, 0, 3);

        #pragma unroll
        for (int kk = 0; kk < KP; kk += 4) {
            v2f a0 = *(const v2f*)(a0B + kk);
            v2f a1 = *(const v2f*)(a1B + kk);
            v2f b0 = *(const v2f*)(b0B + kk);
            v2f b1 = *(const v2f*)(b1B + kk);
            c00 = WMMA_F32(a0, b0, c00);
            c01 = WMMA_F32(a0, b1, c01);
            c10 = WMMA_F32(a1, b0, c10);
            c11 = WMMA_F32(a1, b1, c11);
        }
    }

    const int nCol0 = nBase + wv * 32 + lm;
    const int nCol1 = nCol0 + 16;
    #pragma unroll
    for (int r = 0; r < 8; ++r) {
        int m0 = r + 8 * half;
        if (m0 < valid) {
            size_t p = (size_t)sPair[m0] * H_DIM;        // pair-major output
            out[p + nCol0] = c00[r];
            out[p + nCol1] = c01[r];
        }
        int m1 = m0 + 16;
        if (m1 < valid) {
            size_t p = (size_t)sPair[m1] * H_DIM;
            out[p + nCol0] = c10[r];
            out[p + nCol1] = c11[r];
        }
    }
}

// ---------------- host launch ----------------

extern "C" void kernel_launch(void* const* d_in, const int* in_sizes, int n_in,
                              void* d_out, int out_size, void* d_ws, size_t ws_size,
                              hipStream_t stream) {
    (void)in_sizes; (void)n_in; (void)out_size; (void)ws_size;

    const float* hidden = (const float*)d_in[0];   // (T, H) f32
    const int*   sel    = (const int*)  d_in[1];   // (T, K) i32
    const float* gup    = (const float*)d_in[2];   // (E, 2I, H) f32
    const float* dproj  = (const float*)d_in[3];   // (E, H, I) f32
    float*       out    = (float*)d_out;           // (T, K, H) f32

    char* ws = (char*)d_ws;
    int*   counts  = (int*)(ws + 0);
    int*   offsets = (int*)(ws + 64);
    int*   fill    = (int*)(ws + 128);
    int*   rowsArr = (int*)(ws + 256);                 // P ints
    float* inter   = (float*)(ws + 16384);             // P * I floats (8 MB)

    moe_route_init <<<1, 32, 0, stream>>>(counts, fill);
    moe_route_count<<<P_PAIR / 256, 256, 0, stream>>>(sel, counts);
    moe_route_scan <<<1, 32, 0, stream>>>(counts, offsets);
    moe_route_fill <<<P_PAIR / 256, 256, 0, stream>>>(sel, offsets, fill, rowsArr);

    dim3 g1(P_PAIR / 32, I_DIM / 128, E_EXP);   // (m-tiles, n-blocks, experts)
    moe_gemm1_silu<<<g1, 256, 0, stream>>>(hidden, gup, counts, offsets, rowsArr, inter);

    dim3 g2(P_PAIR / 32, H_DIM / 256, E_EXP);
    moe_gemm2_scatter<<<g2, 256, 0, stream>>>(inter, dproj, counts, offsets, rowsArr, out);
}